// LogitsOnlyTCSLoss_26096221291225
// MI455X (gfx1250) — compile-verified
//
#include <hip/hip_runtime.h>
#include <stdint.h>

#define IGNORE_INDEX (-100)
#define TOPK 100
#define TEMP 5.0f
#define LAMBDA_TCS 10.0f
#define BLOCK 256
#define NWAVE (BLOCK / 32)

typedef float v4f __attribute__((ext_vector_type(4)));

// branchless monotonic float<->uint key mapping
__device__ __forceinline__ uint32_t f2key(uint32_t u) {
    return u ^ (uint32_t)(((int32_t)u >> 31) | 0x80000000);
}
__device__ __forceinline__ float key2f(uint32_t k) {
    uint32_t u = (k & 0x80000000u) ? (k & 0x7FFFFFFFu) : ~k;
    return __uint_as_float(u);
}

__device__ __forceinline__ void osm_upd(float& m, float& s, float x) {
    if (x > m) { s = s * __expf(m - x) + 1.0f; m = x; }
    else       { s += __expf(x - m); }
}

// deterministic wave32 butterfly reductions
__device__ __forceinline__ float waveMaxF(float v) {
#pragma unroll
    for (int off = 16; off > 0; off >>= 1) v = fmaxf(v, __shfl_xor(v, off, 32));
    return v;
}
__device__ __forceinline__ float waveSumF(float v) {
#pragma unroll
    for (int off = 16; off > 0; off >>= 1) v += __shfl_xor(v, off, 32);
    return v;
}
__device__ __forceinline__ void waveMergeOSM(float& m, float& s) {
#pragma unroll
    for (int off = 16; off > 0; off >>= 1) {
        float om = __shfl_xor(m, off, 32);
        float os = __shfl_xor(s, off, 32);
        float M  = fmaxf(m, om);
        s = s * __expf(m - M) + os * __expf(om - M);
        m = M;
    }
}

// One workgroup per token.
__global__ void tcs_token_kernel(const float* __restrict__ student,
                                 const float* __restrict__ teacher,
                                 const int*   __restrict__ labels,
                                 float* __restrict__ wsN,
                                 float* __restrict__ wsK,
                                 float* __restrict__ wsV,
                                 int V) {
    extern __shared__ unsigned char smemRaw[];
    uint32_t* kRow   = (uint32_t*)smemRaw;                  // V keys (async-DMA'd floats, converted in place)
    uint32_t* hist   = (uint32_t*)(smemRaw + (size_t)V * 4);// 256
    int*      wOffG  = (int*)(hist + 256);                  // 8 wave offsets (greater)
    int*      wOffE  = (int*)(wOffG + NWAVE);               // 8 wave offsets (equal)
    float*    redM   = (float*)(wOffE + NWAVE);             // 8 per-wave softmax max
    float*    redS   = (float*)(redM + NWAVE);              // 8 per-wave softmax sum
    int*      topIdx = (int*)(redS + NWAVE);                // 128
    uint32_t* topKey = (uint32_t*)(topIdx + 128);           // 128
    int*      bcast  = (int*)(topKey + 128);                // 8
    float*    fb     = (float*)(bcast + 8);                 // 16

    const int tid  = threadIdx.x;
    const int lane = tid & 31;
    const int wid  = tid >> 5;
    const int tok  = blockIdx.x;
    const float* tBase = teacher + (size_t)tok * V;
    const float* sBase = student + (size_t)tok * V;
    const int nv4 = V >> 2;
    const int K = (TOPK < V) ? TOPK : V;

    // ---- Phase A: async DMA teacher row -> LDS (ASYNCcnt path, NT in cache) ----
    {
        uint32_t ldsBase = (uint32_t)(uintptr_t)kRow;  // low 32 bits = LDS offset
        for (int j = tid; j < nv4; j += BLOCK) {
            uint32_t lds = ldsBase + 16u * (uint32_t)j;
            const float* g = tBase + 4 * j;
            asm volatile("global_load_async_to_lds_b128 %0, %1, off th:TH_LOAD_NT"
                         :: "v"(lds), "v"(g) : "memory");
        }
    }

    // ---- Phase B (overlapped): stream student row (NT), online softmax ----
    {
        float m = -__builtin_huge_valf(), s = 0.0f;
        const v4f* s4 = (const v4f*)sBase;
        for (int j = tid; j < nv4; j += BLOCK) {
            v4f x = __builtin_nontemporal_load(s4 + j);
            osm_upd(m, s, x.x);
            osm_upd(m, s, x.y);
            osm_upd(m, s, x.z);
            osm_upd(m, s, x.w);
        }
        waveMergeOSM(m, s);                 // deterministic butterfly
        if (lane == 0) { redM[wid] = m; redS[wid] = s; }
    }
    if (tid == 0) bcast[0] = K;             // remaining "need" for radix select

    // wait own wave's async DMA, then barrier so all waves' LDS data is complete
    asm volatile("s_wait_asynccnt 0" ::: "memory");
    __syncthreads();

    // ---- convert teacher floats to sortable keys, in place (one sweep) ----
    for (int j = tid; j < V; j += BLOCK) kRow[j] = f2key(kRow[j]);
    __syncthreads();

    // ---- Phase C: radix select (4 x 8-bit digits) for K-th largest key ----
    uint32_t prefix = 0, pmask = 0;
    for (int pass = 0; pass < 4; ++pass) {
        const int shift = 24 - 8 * pass;
        hist[tid] = 0;
        __syncthreads();
        for (int j = tid; j < V; j += BLOCK) {
            uint32_t key = kRow[j];
            if ((key & pmask) == prefix)
                atomicAdd(&hist[(key >> shift) & 255u], 1u);
        }
        __syncthreads();
        if (tid < 32) {   // wave 0: parallel suffix-scan bin selection
            int need = bcast[0];
            unsigned c8 = 0;
#pragma unroll
            for (int b = 0; b < 8; ++b) c8 += hist[8 * tid + b];
            unsigned suf = c8;               // inclusive suffix sum over lanes
#pragma unroll
            for (int off = 1; off < 32; off <<= 1) {
                unsigned t = __shfl_down(suf, off, 32);
                if (tid + off < 32) suf += t;
            }
            unsigned above = suf - c8;       // count in strictly higher groups
            if (suf >= (unsigned)need && above < (unsigned)need) {  // exactly one lane
                unsigned cum = above;
                for (int b = 8 * tid + 7; b >= 8 * tid; --b) {
                    unsigned c = hist[b];
                    if (cum + c >= (unsigned)need) {
                        bcast[0] = need - (int)cum;
                        bcast[1] = b;
                        break;
                    }
                    cum += c;
                }
            }
        }
        __syncthreads();
        prefix |= ((uint32_t)bcast[1]) << shift;
        pmask  |= 0xFFu << shift;
    }
    const uint32_t Tkey = prefix;            // exact key of K-th largest
    const int needFinal = bcast[0];          // ties to take at Tkey
    const int numG = K - needFinal;

    // ---- Phase D: deterministic compaction (rank by position, no atomics) ----
    const int per = V / BLOCK;
    const int rem = V - per * BLOCK;
    const int start = tid * per + (tid < rem ? tid : rem);
    const int cnt   = per + (tid < rem ? 1 : 0);
    int lg = 0, le = 0;
    for (int i = start; i < start + cnt; ++i) {
        uint32_t key = kRow[i];
        lg += (key > Tkey);
        le += (key == Tkey);
    }
    int gi = lg, ei = le;                    // per-wave inclusive scan
#pragma unroll
    for (int off = 1; off < 32; off <<= 1) {
        int tg = __shfl_up(gi, off, 32);
        int te = __shfl_up(ei, off, 32);
        if (lane >= off) { gi += tg; ei += te; }
    }
    if (lane == 31) { wOffG[wid] = gi; wOffE[wid] = ei; }
    __syncthreads();
    if (tid == 0) {                          // exclusive scan of 8 wave totals
        int a = 0, b = 0;
#pragma unroll
        for (int w = 0; w < NWAVE; ++w) {
            int tg = wOffG[w], te = wOffE[w];
            wOffG[w] = a; wOffE[w] = b;
            a += tg; b += te;
        }
    }
    __syncthreads();
    {
        int gpos = wOffG[wid] + (gi - lg);
        int epos = wOffE[wid] + (ei - le);
        for (int i = start; i < start + cnt; ++i) {
            uint32_t key = kRow[i];
            if (key > Tkey) {
                topIdx[gpos] = i; topKey[gpos] = key; ++gpos;
            } else if (key == Tkey && epos < needFinal) {
                topIdx[numG + epos] = i; topKey[numG + epos] = key; ++epos;
            }
        }
    }
    __syncthreads();

    // ---- Phase E: top-K KL, parallel over threads 0..127 (padded) ----
    float tt = -__builtin_huge_valf(), st = -__builtin_huge_valf();
    if (tid < K) {
        tt = key2f(topKey[tid]) * (1.0f / TEMP);
        st = sBase[topIdx[tid]] * (1.0f / TEMP);
    }
    if (tid < 128) {
        float tmw = waveMaxF(tt), smw = waveMaxF(st);
        if (lane == 0) { fb[2 + wid] = tmw; fb[6 + wid] = smw; }
    }
    __syncthreads();
    const float tmax = fmaxf(fmaxf(fb[2], fb[3]), fmaxf(fb[4], fb[5]));
    const float smax = fmaxf(fmaxf(fb[6], fb[7]), fmaxf(fb[8], fb[9]));
    if (tid < 128) {
        float et = __expf(tt - tmax);        // padded lanes: exp(-inf)=0
        float es = __expf(st - smax);
        float tsw = waveSumF(et), ssw = waveSumF(es);
        if (lane == 0) { fb[2 + wid] = tsw; fb[6 + wid] = ssw; }
    }
    __syncthreads();
    const float tsum = ((fb[2] + fb[3]) + (fb[4] + fb[5]));
    const float ssum = ((fb[6] + fb[7]) + (fb[8] + fb[9]));
    const float lts = __logf(tsum), lss = __logf(ssum);
    if (tid < 128) {
        float kli = 0.0f;
        if (tid < K) {
            float lpt = tt - tmax - lts;
            float lps = st - smax - lss;
            kli = __expf(lpt) * (lpt - lps);
        }
        float klw = waveSumF(kli);
        if (lane == 0) fb[2 + wid] = klw;
    }
    __syncthreads();

    // ---- Phase F: per-token outputs (thread 0, tiny fixed-order tail) ----
    if (tid == 0) {
        float kl = ((fb[2] + fb[3]) + (fb[4] + fb[5]));
        float M = -__builtin_huge_valf();
#pragma unroll
        for (int w = 0; w < NWAVE; ++w) M = fmaxf(M, redM[w]);
        float S = 0.0f;
#pragma unroll
        for (int w = 0; w < NWAVE; ++w) S += redS[w] * __expf(redM[w] - M);
        float logS = __logf(S);

        int   label = labels[tok];
        bool  valid = (label != IGNORE_INDEX);
        int   lab   = valid ? label : 0;
        float xl    = sBase[lab];
        float nll   = -(xl - M - logS);

        wsN[tok] = valid ? nll : 0.0f;
        wsK[tok] = valid ? kl  : 0.0f;
        wsV[tok] = valid ? 1.0f : 0.0f;
    }
}

__global__ void tcs_reduce_kernel(const float* __restrict__ wsN,
                                  const float* __restrict__ wsK,
                                  const float* __restrict__ wsV,
                                  float* __restrict__ out, int T) {
    __shared__ float rn[BLOCK], rk[BLOCK], rv[BLOCK];
    const int tid = threadIdx.x;
    float n = 0.0f, k = 0.0f, v = 0.0f;
    for (int i = tid; i < T; i += BLOCK) { n += wsN[i]; k += wsK[i]; v += wsV[i]; }
    rn[tid] = n; rk[tid] = k; rv[tid] = v;
    __syncthreads();
    for (int off = BLOCK / 2; off > 0; off >>= 1) {
        if (tid < off) {
            rn[tid] += rn[tid + off];
            rk[tid] += rk[tid + off];
            rv[tid] += rv[tid + off];
        }
        __syncthreads();
    }
    if (tid == 0) {
        float nv  = fmaxf(rv[0], 1.0f);
        float ce  = rn[0] / nv;
        float tcs = rk[0] / nv * (TEMP * TEMP);
        out[0] = ce + LAMBDA_TCS * tcs;
        out[1] = ce;
        out[2] = tcs;
        out[3] = 0.0f;
    }
}

extern "C" void kernel_launch(void* const* d_in, const int* in_sizes, int n_in,
                              void* d_out, int out_size, void* d_ws, size_t ws_size,
                              hipStream_t stream) {
    const float* student = (const float*)d_in[0];
    const float* teacher = (const float*)d_in[1];
    const int*   labels  = (const int*)d_in[2];   // harness contract: integer -> const int*

    const int T = in_sizes[2];            // B*S tokens
    const int V = in_sizes[0] / T;        // vocab size

    float* wsN = (float*)d_ws;
    float* wsK = wsN + T;
    float* wsV = wsK + T;

    size_t smemBytes = (size_t)V * 4 + 4096;

    tcs_token_kernel<<<T, BLOCK, smemBytes, stream>>>(student, teacher, labels,
                                                      wsN, wsK, wsV, V);
    tcs_reduce_kernel<<<1, BLOCK, 0, stream>>>(wsN, wsK, wsV, (float*)d_out, T);
}